// HMLDM_37065567765089
// MI455X (gfx1250) — compile-verified
//
#include <hip/hip_runtime.h>
#include <hip/hip_bf16.h>

typedef __attribute__((ext_vector_type(2))) float v2f;
typedef __attribute__((ext_vector_type(8))) float v8f;

// Problem constants (match reference)
constexpr int NR = 8192;
constexpr int NC = 8192;
constexpr int D  = 16;
constexpr int N_EDGES = 1048576;

constexpr int ROW_TILES      = NR / 16;                         // 512
constexpr int COL_TILES      = NC / 16;                         // 512
constexpr int TILES_PER_WAVE = 8;                               // col tiles per wave
constexpr int COL_GROUPS     = COL_TILES / TILES_PER_WAVE;      // 64
constexpr int WAVES_PER_BLK  = 8;                               // 256 threads / wave32
constexpr int PAIR_BLOCKS    = ROW_TILES * COL_GROUPS / WAVES_PER_BLK; // 4096
constexpr int EDGE_BLOCKS    = 1024;

constexpr float LOG2E = 1.4426950408889634f;   // log2(e)

// Fast hardware sqrt: inputs are clamped >= 1e-12 (~2^-40), far above the
// 2^-96 range where v_sqrt_f32 needs the denormal fixup the compiler would
// otherwise emit for sqrtf().
__device__ __forceinline__ float fast_sqrtf(float x) {
    return __builtin_amdgcn_sqrtf(x);
}

// ---------------------------------------------------------------------------
// Prep: row softmax of latent[i,0:16], sum-of-squares, exp(gamma)
// ---------------------------------------------------------------------------
__global__ __launch_bounds__(256)
void prep_kernel(const float* __restrict__ lat, const float* __restrict__ gamma,
                 float* __restrict__ zout, float* __restrict__ sqout,
                 float* __restrict__ egout, int n) {
    int i = blockIdx.x * blockDim.x + threadIdx.x;
    if (i >= n) return;
    float v[16];
    const float4* src = (const float4*)(lat + (size_t)i * D);
#pragma unroll
    for (int k = 0; k < 4; ++k) {
        float4 t = src[k];
        v[4*k+0] = t.x; v[4*k+1] = t.y; v[4*k+2] = t.z; v[4*k+3] = t.w;
    }
    float mx = v[0];
#pragma unroll
    for (int d = 1; d < 16; ++d) mx = fmaxf(mx, v[d]);
    float s = 0.f;
#pragma unroll
    for (int d = 0; d < 16; ++d) { v[d] = __expf(v[d] - mx); s += v[d]; }
    float inv = 1.f / s;
    float ss = 0.f;
#pragma unroll
    for (int d = 0; d < 16; ++d) { v[d] *= inv; ss += v[d] * v[d]; }
    float4* dst = (float4*)(zout + (size_t)i * D);
#pragma unroll
    for (int k = 0; k < 4; ++k) {
        float4 t; t.x = v[4*k+0]; t.y = v[4*k+1]; t.z = v[4*k+2]; t.w = v[4*k+3];
        dst[k] = t;
    }
    sqout[i] = ss;
    egout[i] = __expf(gamma[i]);
}

// ---------------------------------------------------------------------------
// Pairwise term: one wave per (row_tile, 8 col tiles).
// dot(z_n, w_m) via V_WMMA_F32_16X16X4_F32 x4 (K=16), then
// sum a[n]*b[m]*exp(-(dist+1e-6)) with dist = sqrt(max(zz+ww-2dot,1e-12)).
// exp(-(d+eps)) computed as exp2(fma(d, -log2e, -eps*log2e)): 1 FMA + 1 trans.
// ---------------------------------------------------------------------------
__global__ __launch_bounds__(256)
void pair_kernel(const float* __restrict__ z,  const float* __restrict__ w,
                 const float* __restrict__ zz, const float* __restrict__ ww,
                 const float* __restrict__ ea, const float* __restrict__ eb,
                 float* __restrict__ partials) {
    const int lane = threadIdx.x & 31;
    const int wave = threadIdx.x >> 5;
    const int gwid = blockIdx.x * WAVES_PER_BLK + wave;
    const int row_tile = gwid / COL_GROUPS;
    const int col_grp  = gwid % COL_GROUPS;
    const int row0 = row_tile * 16;
    const int half = lane >> 4;   // 0: lanes 0-15, 1: lanes 16-31
    const int mn   = lane & 15;

    // A operand: lane holds row (row0+mn); K pairs (k0+2*half, k0+2*half+1)
    const v2f* zrow = (const v2f*)(z + (size_t)(row0 + mn) * D);
    v2f A0 = zrow[0 + half];
    v2f A1 = zrow[2 + half];
    v2f A2 = zrow[4 + half];
    v2f A3 = zrow[6 + half];

    // C/D layout: element v is row (v + 8*half), col mn
    float zzr[8], ear[8];
    const float* zzp = zz + row0 + 8 * half;
    const float* eap = ea + row0 + 8 * half;
#pragma unroll
    for (int v = 0; v < 8; ++v) { zzr[v] = zzp[v]; ear[v] = eap[v]; }

    const float eps_l2 = -1e-6f * LOG2E;   // exp(-(d+eps)) = exp2(-d*log2e - eps*log2e)

    float lane_acc = 0.f;

    for (int t = 0; t < TILES_PER_WAVE; ++t) {
        const int col0 = (col_grp * TILES_PER_WAVE + t) * 16;
        const v2f* wrow = (const v2f*)(w + (size_t)(col0 + mn) * D);
        v2f B0 = wrow[0 + half];
        v2f B1 = wrow[2 + half];
        v2f B2 = wrow[4 + half];
        v2f B3 = wrow[6 + half];

        v8f C = {};
        C = __builtin_amdgcn_wmma_f32_16x16x4_f32(false, A0, false, B0, (short)0, C, false, false);
        C = __builtin_amdgcn_wmma_f32_16x16x4_f32(false, A1, false, B1, (short)0, C, false, false);
        C = __builtin_amdgcn_wmma_f32_16x16x4_f32(false, A2, false, B2, (short)0, C, false, false);
        C = __builtin_amdgcn_wmma_f32_16x16x4_f32(false, A3, false, B3, (short)0, C, false, false);

        const float wwn = ww[col0 + mn];
        const float ebn = eb[col0 + mn];
        float ts = 0.f;
#pragma unroll
        for (int v = 0; v < 8; ++v) {
            float sq = zzr[v] + wwn - 2.f * C[v];
            float d  = fast_sqrtf(fmaxf(sq, 1e-12f));
            float e  = __builtin_amdgcn_exp2f(fmaf(d, -LOG2E, eps_l2));
            ts = fmaf(ear[v], e, ts);
        }
        lane_acc = fmaf(ebn, ts, lane_acc);
    }

    // wave32 reduction
    float s = lane_acc;
#pragma unroll
    for (int off = 16; off > 0; off >>= 1) s += __shfl_xor(s, off, 32);

    __shared__ float wsum[WAVES_PER_BLK];
    if (lane == 0) wsum[wave] = s;
    __syncthreads();
    if (threadIdx.x == 0) {
        float b = 0.f;
#pragma unroll
        for (int i = 0; i < WAVES_PER_BLK; ++i) b += wsum[i];
        partials[blockIdx.x] = b;
    }
}

// ---------------------------------------------------------------------------
// Edge term: gather z[r], w[c] (int64 indices), sum gw*(gr+gc-||z-w||)
// ---------------------------------------------------------------------------
__global__ __launch_bounds__(256)
void edge_kernel(const float* __restrict__ z,  const float* __restrict__ w,
                 const float* __restrict__ gr, const float* __restrict__ gc,
                 const float* __restrict__ gwt,
                 const long long* __restrict__ ri, const long long* __restrict__ ci,
                 float* __restrict__ partials, int nE) {
    int tid = blockIdx.x * blockDim.x + threadIdx.x;
    int stride = gridDim.x * blockDim.x;
    float acc = 0.f;
    for (int e = tid; e < nE; e += stride) {
        int r = (int)ri[e];
        int c = (int)ci[e];
        const float4* zp = (const float4*)(z + (size_t)r * D);
        const float4* wp = (const float4*)(w + (size_t)c * D);
        float s = 0.f;
#pragma unroll
        for (int k = 0; k < 4; ++k) {
            float4 a4 = zp[k], b4 = wp[k];
            float d0 = a4.x - b4.x, d1 = a4.y - b4.y;
            float d2 = a4.z - b4.z, d3 = a4.w - b4.w;
            s += d0*d0 + d1*d1 + d2*d2 + d3*d3;
        }
        float dl = fast_sqrtf(fmaxf(s, 1e-12f));
        acc += gwt[e] * (gr[r] + gc[c] - dl);   // delta=1, p=1
    }
    for (int off = 16; off > 0; off >>= 1) acc += __shfl_xor(acc, off, 32);
    __shared__ float wsum[WAVES_PER_BLK];
    int lane = threadIdx.x & 31, wv = threadIdx.x >> 5;
    if (lane == 0) wsum[wv] = acc;
    __syncthreads();
    if (threadIdx.x == 0) {
        float b = 0.f;
#pragma unroll
        for (int i = 0; i < WAVES_PER_BLK; ++i) b += wsum[i];
        partials[blockIdx.x] = b;
    }
}

// ---------------------------------------------------------------------------
// Finalize: out = pair_sum - edge_sum (double accumulation, fixed order)
// ---------------------------------------------------------------------------
__global__ __launch_bounds__(256)
void finalize_kernel(const float* __restrict__ pairp, int np,
                     const float* __restrict__ edgep, int ne,
                     float* __restrict__ out) {
    __shared__ double sh[256];
    double s1 = 0.0, s2 = 0.0;
    for (int i = threadIdx.x; i < np; i += 256) s1 += (double)pairp[i];
    for (int i = threadIdx.x; i < ne; i += 256) s2 += (double)edgep[i];
    sh[threadIdx.x] = s1 - s2;
    __syncthreads();
    for (int off = 128; off > 0; off >>= 1) {
        if (threadIdx.x < off) sh[threadIdx.x] += sh[threadIdx.x + off];
        __syncthreads();
    }
    if (threadIdx.x == 0) out[0] = (float)sh[0];
}

extern "C" void kernel_launch(void* const* d_in, const int* in_sizes, int n_in,
                              void* d_out, int out_size, void* d_ws, size_t ws_size,
                              hipStream_t stream) {
    (void)in_sizes; (void)n_in; (void)out_size; (void)ws_size;
    const float*     gamma_rows = (const float*)d_in[0];
    const float*     gamma_cols = (const float*)d_in[1];
    const float*     latent_z   = (const float*)d_in[2];
    const float*     latent_w   = (const float*)d_in[3];
    const float*     graph_w    = (const float*)d_in[4];
    const long long* rows_idx   = (const long long*)d_in[5];
    const long long* col_idx    = (const long long*)d_in[6];
    float* out = (float*)d_out;

    // Workspace layout (floats)
    float* ws  = (float*)d_ws;
    float* z   = ws;                 // NR*D
    float* w   = z  + (size_t)NR * D;// NC*D
    float* zz  = w  + (size_t)NC * D;// NR
    float* ww  = zz + NR;            // NC
    float* ea  = ww + NC;            // NR
    float* eb  = ea + NR;            // NC
    float* pairp = eb + NC;          // PAIR_BLOCKS
    float* edgep = pairp + PAIR_BLOCKS; // EDGE_BLOCKS

    prep_kernel<<<NR / 256, 256, 0, stream>>>(latent_z, gamma_rows, z, zz, ea, NR);
    prep_kernel<<<NC / 256, 256, 0, stream>>>(latent_w, gamma_cols, w, ww, eb, NC);
    pair_kernel<<<PAIR_BLOCKS, 256, 0, stream>>>(z, w, zz, ww, ea, eb, pairp);
    edge_kernel<<<EDGE_BLOCKS, 256, 0, stream>>>(z, w, gamma_rows, gamma_cols,
                                                 graph_w, rows_idx, col_idx,
                                                 edgep, N_EDGES);
    finalize_kernel<<<1, 256, 0, stream>>>(pairp, PAIR_BLOCKS, edgep, EDGE_BLOCKS, out);
}